// SSPQTokenizer_10763188044459
// MI455X (gfx1250) — compile-verified
//
#include <hip/hip_runtime.h>
#include <math.h>

// ---------------------------------------------------------------------------
// SSPQ tokenizer for MI455X (gfx1250, wave32).
// Dominant op: two [N,64]x[64,1024] f32 distance matmuls -> V_WMMA_F32_16X16X4_F32.
// ---------------------------------------------------------------------------

#define DVQ 64
#define KCB 1024
#define BETA 0.25f

typedef float f2_t __attribute__((ext_vector_type(2)));
typedef float f8_t __attribute__((ext_vector_type(8)));

__device__ __forceinline__ float gelu_f(float x) {
  return 0.5f * x * (1.0f + erff(x * 0.70710678118654752440f));
}

// ---------------------------------------------------------------------------
// prep: codebook squared norms, zero histogram counters and loss accumulators
// ---------------------------------------------------------------------------
__global__ void __launch_bounds__(256) k_prep(
    const float* __restrict__ cb_s, const float* __restrict__ cb_x,
    float* __restrict__ cnorm_s, float* __restrict__ cnorm_x,
    float* __restrict__ counts_s, float* __restrict__ counts_x,
    float* __restrict__ accums) {
  int tid = blockIdx.x * 256 + threadIdx.x;
  if (tid < KCB) {
    const float* p = cb_s + (size_t)tid * DVQ;
    float s = 0.f;
#pragma unroll 8
    for (int i = 0; i < DVQ; i++) s += p[i] * p[i];
    cnorm_s[tid] = s;
    counts_s[tid] = 0.f;
  } else if (tid < 2 * KCB) {
    int k = tid - KCB;
    const float* p = cb_x + (size_t)k * DVQ;
    float s = 0.f;
#pragma unroll 8
    for (int i = 0; i < DVQ; i++) s += p[i] * p[i];
    cnorm_x[k] = s;
    counts_x[k] = 0.f;
  }
  if (tid < 4) accums[tid] = 0.f;
}

// ---------------------------------------------------------------------------
// Spectral encoder: spec mean -> conv1d(1->16) gelu -> conv1d(16->16) gelu ->
// FC 128->64.  256 threads = 4 tokens, 64 threads per token.
// ---------------------------------------------------------------------------
__global__ void __launch_bounds__(256) k_spec_encode(
    const float* __restrict__ cubes,
    const float* __restrict__ w1, const float* __restrict__ b1,
    const float* __restrict__ w2, const float* __restrict__ b2,
    const float* __restrict__ fcw, const float* __restrict__ fcb,
    float* __restrict__ ze) {
  __shared__ float sw1[48], sb1[16], sw2[768], sb2[16];
  __shared__ float spec[4][8];
  __shared__ float s1[4][128];
  __shared__ float s2[4][128];
  const int t = threadIdx.x, slot = t >> 6, lt = t & 63;
  const int tok = blockIdx.x * 4 + slot;
  for (int i = t; i < 48; i += 256) sw1[i] = w1[i];
  if (t < 16) { sb1[t] = b1[t]; sb2[t] = b2[t]; }
  for (int i = t; i < 768; i += 256) sw2[i] = w2[i];
  __syncthreads();
  if (lt < 8) {
    const float* p = cubes + (size_t)tok * 72 + lt * 9;
    float s = 0.f;
#pragma unroll
    for (int i = 0; i < 9; i++) s += p[i];
    spec[slot][lt] = s * (1.f / 9.f);
  }
  __syncthreads();
#pragma unroll
  for (int r = 0; r < 2; r++) {
    int o = lt + 64 * r, c = o >> 3, l = o & 7;
    float a = sb1[c];
#pragma unroll
    for (int k = 0; k < 3; k++) {
      int ll = l + k - 1;
      if (ll >= 0 && ll < 8) a += sw1[c * 3 + k] * spec[slot][ll];
    }
    s1[slot][o] = gelu_f(a);
  }
  __syncthreads();
#pragma unroll
  for (int r = 0; r < 2; r++) {
    int o = lt + 64 * r, c = o >> 3, l = o & 7;
    float a = sb2[c];
    for (int ci = 0; ci < 16; ci++) {
#pragma unroll
      for (int k = 0; k < 3; k++) {
        int ll = l + k - 1;
        if (ll >= 0 && ll < 8) a += sw2[(c * 16 + ci) * 3 + k] * s1[slot][ci * 8 + ll];
      }
    }
    s2[slot][o] = gelu_f(a);
  }
  __syncthreads();
  {
    float a = fcb[lt];
    const float* w = fcw + (size_t)lt * 128;
    const float* sp = s2[slot];
#pragma unroll 8
    for (int j = 0; j < 128; j++) a += w[j] * sp[j];
    ze[(size_t)tok * DVQ + lt] = a;
  }
}

// ---------------------------------------------------------------------------
// Spatial encoder: conv2d(8->16) gelu -> conv2d(16->16) gelu -> FC 144->64.
// ---------------------------------------------------------------------------
__global__ void __launch_bounds__(256) k_spat_encode(
    const float* __restrict__ cubes,
    const float* __restrict__ w1, const float* __restrict__ b1,
    const float* __restrict__ w2, const float* __restrict__ b2,
    const float* __restrict__ fcw, const float* __restrict__ fcb,
    float* __restrict__ ze) {
  __shared__ float sw1[1152], sb1[16], sw2[2304], sb2[16];
  __shared__ float in[4][72];
  __shared__ float x1[4][144];
  __shared__ float x2[4][144];
  const int t = threadIdx.x, slot = t >> 6, lt = t & 63;
  const int tok = blockIdx.x * 4 + slot;
  for (int i = t; i < 1152; i += 256) sw1[i] = w1[i];
  for (int i = t; i < 2304; i += 256) sw2[i] = w2[i];
  if (t < 16) { sb1[t] = b1[t]; sb2[t] = b2[t]; }
  __syncthreads();
  {
    const float* p = cubes + (size_t)tok * 72;
    in[slot][lt] = p[lt];
    if (lt < 8) in[slot][lt + 64] = p[lt + 64];
  }
  __syncthreads();
#pragma unroll
  for (int r = 0; r < 3; r++) {
    int o = lt + 64 * r;
    if (o < 144) {
      int c = o / 9, ij = o % 9, i = ij / 3, j = ij % 3;
      float a = sb1[c];
      for (int ci = 0; ci < 8; ci++)
#pragma unroll
        for (int ki = 0; ki < 3; ki++)
#pragma unroll
          for (int kj = 0; kj < 3; kj++) {
            int ii = i + ki - 1, jj = j + kj - 1;
            if (ii >= 0 && ii < 3 && jj >= 0 && jj < 3)
              a += sw1[((c * 8 + ci) * 3 + ki) * 3 + kj] * in[slot][ci * 9 + ii * 3 + jj];
          }
      x1[slot][o] = gelu_f(a);
    }
  }
  __syncthreads();
#pragma unroll
  for (int r = 0; r < 3; r++) {
    int o = lt + 64 * r;
    if (o < 144) {
      int c = o / 9, ij = o % 9, i = ij / 3, j = ij % 3;
      float a = sb2[c];
      for (int ci = 0; ci < 16; ci++)
#pragma unroll
        for (int ki = 0; ki < 3; ki++)
#pragma unroll
          for (int kj = 0; kj < 3; kj++) {
            int ii = i + ki - 1, jj = j + kj - 1;
            if (ii >= 0 && ii < 3 && jj >= 0 && jj < 3)
              a += sw2[((c * 16 + ci) * 3 + ki) * 3 + kj] * x1[slot][ci * 9 + ii * 3 + jj];
          }
      x2[slot][o] = gelu_f(a);
    }
  }
  __syncthreads();
  {
    float a = fcb[lt];
    const float* w = fcw + (size_t)lt * 144;
    const float* sp = x2[slot];
#pragma unroll 8
    for (int j = 0; j < 144; j++) a += w[j] * sp[j];
    ze[(size_t)tok * DVQ + lt] = a;
  }
}

// ---------------------------------------------------------------------------
// VQ argmin via V_WMMA_F32_16X16X4_F32.
// 128 threads = 4 waves; each wave owns a 16-token tile and sweeps 64 k-tiles
// of 16 codes, accumulating z.c with 16 f32 WMMAs (K chunks of 4), tracking a
// running min of (|c|^2 - 2 z.c), then a 16-lane shuffle argmin reduce.
// Fragment layouts per ISA: A 16x4 f32: lanes0-15 hold K0,K1 (M=lane),
// lanes16-31 hold K2,K3. B is the mirror (lanes hold code columns), so a
// row-major codebook row loads directly. C/D: acc[r] <-> (M=r+8*hi, N=lane&15).
// ---------------------------------------------------------------------------
__global__ void __launch_bounds__(128) k_vq_argmin(
    const float* __restrict__ Z, const float* __restrict__ CB,
    const float* __restrict__ cnorm, int* __restrict__ out_idx) {
  const int wave = threadIdx.x >> 5;
  const int lane = threadIdx.x & 31;
  const int tile = blockIdx.x * 4 + wave;
  const int m0 = tile * 16;
  const int col = lane & 15;
  const int hi = lane >> 4;
  const int koff = hi * 2;

  f2_t a[16];
  {
    const float* zrow = Z + (size_t)(m0 + col) * DVQ + koff;
#pragma unroll
    for (int q = 0; q < 16; q++) { a[q].x = zrow[4 * q]; a[q].y = zrow[4 * q + 1]; }
  }

  float rmin[8];
  int ridx[8];
#pragma unroll
  for (int r = 0; r < 8; r++) { rmin[r] = 1e30f; ridx[r] = 0; }

  for (int kt = 0; kt < 64; kt++) {
    const int n = kt * 16 + col;
    const float* crow = CB + (size_t)n * DVQ + koff;
    f2_t bf[16];
#pragma unroll
    for (int q = 0; q < 16; q++) { bf[q].x = crow[4 * q]; bf[q].y = crow[4 * q + 1]; }
    f8_t acc = {0.f, 0.f, 0.f, 0.f, 0.f, 0.f, 0.f, 0.f};
#pragma unroll
    for (int q = 0; q < 16; q++)
      acc = __builtin_amdgcn_wmma_f32_16x16x4_f32(false, a[q], false, bf[q],
                                                  (short)0, acc, false, false);
    const float cn = cnorm[n];
#pragma unroll
    for (int r = 0; r < 8; r++) {
      float s = cn - 2.0f * acc[r];
      if (s < rmin[r]) { rmin[r] = s; ridx[r] = n; }
    }
  }

  // argmin reduce across the 16 lanes of each half-wave (prefer smaller index)
#pragma unroll
  for (int r = 0; r < 8; r++) {
    float m = rmin[r];
    int i = ridx[r];
    for (int off = 8; off > 0; off >>= 1) {
      float om = __shfl_xor(m, off, 16);
      int oi = __shfl_xor(i, off, 16);
      if (om < m || (om == m && oi < i)) { m = om; i = oi; }
    }
    rmin[r] = m;
    ridx[r] = i;
  }
  if (col == 0) {
#pragma unroll
    for (int r = 0; r < 8; r++) out_idx[m0 + r + 8 * hi] = ridx[r];
  }
}

// ---------------------------------------------------------------------------
// finish: gather z_q = cb[idx] -> emb output (straight-through forward value),
// write float(idx), histogram count, accumulate sum (z_e - z_q)^2.
// ---------------------------------------------------------------------------
__global__ void __launch_bounds__(256) k_finish_vq(
    const float* __restrict__ ze, const float* __restrict__ cb,
    const int* __restrict__ idx, float* __restrict__ emb_out,
    float* __restrict__ kidx_out, float* __restrict__ counts,
    float* __restrict__ accum) {
  __shared__ float red[256];
  const int t = threadIdx.x, slot = t >> 6, lt = t & 63;
  const int tok = blockIdx.x * 4 + slot;
  const int k = idx[tok];
  const float zq = cb[(size_t)k * DVQ + lt];
  const float zev = ze[(size_t)tok * DVQ + lt];
  const float d = zev - zq;
  emb_out[(size_t)tok * DVQ + lt] = zq;
  if (lt == 0) {
    kidx_out[tok] = (float)k;
    atomicAdd(&counts[k], 1.0f);
  }
  red[t] = d * d;
  __syncthreads();
  for (int s = 128; s > 0; s >>= 1) {
    if (t < s) red[t] += red[t + s];
    __syncthreads();
  }
  if (t == 0) atomicAdd(accum, red[0]);
}

// ---------------------------------------------------------------------------
// Spectral decoder + recon loss: FC 64->128 -> conv1d(16->16) gelu ->
// conv1d(16->1); compare to recomputed spec mean.
// ---------------------------------------------------------------------------
__global__ void __launch_bounds__(256) k_spec_decode(
    const float* __restrict__ zq_all, const float* __restrict__ cubes,
    const float* __restrict__ fcw, const float* __restrict__ fcb,
    const float* __restrict__ w1, const float* __restrict__ b1,
    const float* __restrict__ w2, const float* __restrict__ b2,
    float* __restrict__ accum) {
  __shared__ float dw1[768], db1[16], dw2[48];
  __shared__ float h1[4][128];
  __shared__ float g[4][128];
  __shared__ float red[256];
  const int t = threadIdx.x, slot = t >> 6, lt = t & 63;
  const int tok = blockIdx.x * 4 + slot;
  for (int i = t; i < 768; i += 256) dw1[i] = w1[i];
  if (t < 16) db1[t] = b1[t];
  if (t < 48) dw2[t] = w2[t];
  __syncthreads();
  const float* zq = zq_all + (size_t)tok * DVQ;
#pragma unroll
  for (int r = 0; r < 2; r++) {
    int j = lt + 64 * r;
    float a = fcb[j];
    const float* w = fcw + (size_t)j * DVQ;
#pragma unroll 8
    for (int d = 0; d < DVQ; d++) a += w[d] * zq[d];
    h1[slot][j] = a;
  }
  __syncthreads();
#pragma unroll
  for (int r = 0; r < 2; r++) {
    int o = lt + 64 * r, c = o >> 3, l = o & 7;
    float a = db1[c];
    for (int ci = 0; ci < 16; ci++)
#pragma unroll
      for (int k = 0; k < 3; k++) {
        int ll = l + k - 1;
        if (ll >= 0 && ll < 8) a += dw1[(c * 16 + ci) * 3 + k] * h1[slot][ci * 8 + ll];
      }
    g[slot][o] = gelu_f(a);
  }
  __syncthreads();
  float contrib = 0.f;
  if (lt < 8) {
    const int l = lt;
    float a = b2[0];
    for (int ci = 0; ci < 16; ci++)
#pragma unroll
      for (int k = 0; k < 3; k++) {
        int ll = l + k - 1;
        if (ll >= 0 && ll < 8) a += dw2[ci * 3 + k] * g[slot][ci * 8 + ll];
      }
    const float* p = cubes + (size_t)tok * 72 + l * 9;
    float sp = 0.f;
#pragma unroll
    for (int i = 0; i < 9; i++) sp += p[i];
    sp *= (1.f / 9.f);
    float d = a - sp;
    contrib = d * d;
  }
  red[t] = contrib;
  __syncthreads();
  for (int s = 128; s > 0; s >>= 1) {
    if (t < s) red[t] += red[t + s];
    __syncthreads();
  }
  if (t == 0) atomicAdd(accum, red[0]);
}

// ---------------------------------------------------------------------------
// Spatial decoder + recon loss: FC 64->144 -> conv2d(16->16) gelu ->
// conv2d(16->8); compare to raw cube channels.
// ---------------------------------------------------------------------------
__global__ void __launch_bounds__(256) k_spat_decode(
    const float* __restrict__ zq_all, const float* __restrict__ cubes,
    const float* __restrict__ fcw, const float* __restrict__ fcb,
    const float* __restrict__ w1, const float* __restrict__ b1,
    const float* __restrict__ w2, const float* __restrict__ b2,
    float* __restrict__ accum) {
  __shared__ float dw1[2304], db1[16], dw2[1152], db2[8];
  __shared__ float h1[4][144];
  __shared__ float g[4][144];
  __shared__ float red[256];
  const int t = threadIdx.x, slot = t >> 6, lt = t & 63;
  const int tok = blockIdx.x * 4 + slot;
  for (int i = t; i < 2304; i += 256) dw1[i] = w1[i];
  for (int i = t; i < 1152; i += 256) dw2[i] = w2[i];
  if (t < 16) db1[t] = b1[t];
  if (t < 8) db2[t] = b2[t];
  __syncthreads();
  const float* zq = zq_all + (size_t)tok * DVQ;
#pragma unroll
  for (int r = 0; r < 3; r++) {
    int j = lt + 64 * r;
    if (j < 144) {
      float a = fcb[j];
      const float* w = fcw + (size_t)j * DVQ;
#pragma unroll 8
      for (int d = 0; d < DVQ; d++) a += w[d] * zq[d];
      h1[slot][j] = a;
    }
  }
  __syncthreads();
#pragma unroll
  for (int r = 0; r < 3; r++) {
    int o = lt + 64 * r;
    if (o < 144) {
      int c = o / 9, ij = o % 9, i = ij / 3, j = ij % 3;
      float a = db1[c];
      for (int ci = 0; ci < 16; ci++)
#pragma unroll
        for (int ki = 0; ki < 3; ki++)
#pragma unroll
          for (int kj = 0; kj < 3; kj++) {
            int ii = i + ki - 1, jj = j + kj - 1;
            if (ii >= 0 && ii < 3 && jj >= 0 && jj < 3)
              a += dw1[((c * 16 + ci) * 3 + ki) * 3 + kj] * h1[slot][ci * 9 + ii * 3 + jj];
          }
      g[slot][o] = gelu_f(a);
    }
  }
  __syncthreads();
  float contrib = 0.f;
#pragma unroll
  for (int r = 0; r < 2; r++) {
    int o = lt + 64 * r;
    if (o < 72) {
      int ch = o / 9, ij = o % 9, i = ij / 3, j = ij % 3;
      float a = db2[ch];
      for (int ci = 0; ci < 16; ci++)
#pragma unroll
        for (int ki = 0; ki < 3; ki++)
#pragma unroll
          for (int kj = 0; kj < 3; kj++) {
            int ii = i + ki - 1, jj = j + kj - 1;
            if (ii >= 0 && ii < 3 && jj >= 0 && jj < 3)
              a += dw2[((ch * 16 + ci) * 3 + ki) * 3 + kj] * g[slot][ci * 9 + ii * 3 + jj];
          }
      float d = a - cubes[(size_t)tok * 72 + o];
      contrib += d * d;
    }
  }
  red[t] = contrib;
  __syncthreads();
  for (int s = 128; s > 0; s >>= 1) {
    if (t < s) red[t] += red[t + s];
    __syncthreads();
  }
  if (t == 0) atomicAdd(accum, red[0]);
}

// ---------------------------------------------------------------------------
// finalize: perplexities + scalar losses (return_recon read device-side).
// ---------------------------------------------------------------------------
__global__ void __launch_bounds__(256) k_finalize(
    const float* __restrict__ counts_s, const float* __restrict__ counts_x,
    const float* __restrict__ accums, const int* __restrict__ ret_recon,
    float* __restrict__ out_scal, int N) {
  __shared__ float rs[256], rx[256];
  const int t = threadIdx.x;
  float ss = 0.f, sx = 0.f;
  const float invN = 1.0f / (float)N;
  for (int i = t; i < KCB; i += 256) {
    float ps = counts_s[i] * invN;
    ss += ps * logf(ps + 1e-10f);
    float px = counts_x[i] * invN;
    sx += px * logf(px + 1e-10f);
  }
  rs[t] = ss;
  rx[t] = sx;
  __syncthreads();
  for (int s = 128; s > 0; s >>= 1) {
    if (t < s) { rs[t] += rs[t + s]; rx[t] += rx[t + s]; }
    __syncthreads();
  }
  if (t == 0) {
    float invND = 1.0f / ((float)N * (float)DVQ);
    out_scal[0] = BETA * accums[0] * invND + BETA * accums[1] * invND;  // total
    out_scal[1] = expf(-rs[0]);                                        // perplex_s
    out_scal[2] = expf(-rx[0]);                                        // perplex_x
    float rec = accums[2] / ((float)N * 8.f) + accums[3] / ((float)N * 72.f);
    out_scal[3] = (ret_recon[0] != 0) ? rec : 0.0f;
  }
}

// ---------------------------------------------------------------------------
extern "C" void kernel_launch(void* const* d_in, const int* in_sizes, int n_in,
                              void* d_out, int out_size, void* d_ws, size_t ws_size,
                              hipStream_t stream) {
  const float* cubes      = (const float*)d_in[0];
  const float* s_w1       = (const float*)d_in[1];
  const float* s_b1       = (const float*)d_in[2];
  const float* s_w2       = (const float*)d_in[3];
  const float* s_b2       = (const float*)d_in[4];
  const float* s_fc_w     = (const float*)d_in[5];
  const float* s_fc_b     = (const float*)d_in[6];
  const float* cb_s       = (const float*)d_in[7];
  const float* s_dec_fc_w = (const float*)d_in[8];
  const float* s_dec_fc_b = (const float*)d_in[9];
  const float* s_dec_w1   = (const float*)d_in[10];
  const float* s_dec_b1   = (const float*)d_in[11];
  const float* s_dec_w2   = (const float*)d_in[12];
  const float* s_dec_b2   = (const float*)d_in[13];
  const float* x_w1       = (const float*)d_in[14];
  const float* x_b1       = (const float*)d_in[15];
  const float* x_w2       = (const float*)d_in[16];
  const float* x_b2       = (const float*)d_in[17];
  const float* x_fc_w     = (const float*)d_in[18];
  const float* x_fc_b     = (const float*)d_in[19];
  const float* cb_x       = (const float*)d_in[20];
  const float* x_dec_fc_w = (const float*)d_in[21];
  const float* x_dec_fc_b = (const float*)d_in[22];
  const float* x_dec_w1   = (const float*)d_in[23];
  const float* x_dec_b1   = (const float*)d_in[24];
  const float* x_dec_w2   = (const float*)d_in[25];
  const float* x_dec_b2   = (const float*)d_in[26];
  const int*   ret_recon  = (const int*)d_in[27];

  const int N = in_sizes[0] / 72;  // 73728

  // workspace carving (floats)
  float* ws = (float*)d_ws;
  float* zes    = ws;                          // N*64
  float* zex    = zes + (size_t)N * DVQ;       // N*64
  float* cns    = zex + (size_t)N * DVQ;       // 1024
  float* cnx    = cns + KCB;                   // 1024
  float* cts    = cnx + KCB;                   // 1024
  float* ctx    = cts + KCB;                   // 1024
  float* accums = ctx + KCB;                   // 4 (+pad)
  int*   idxs   = (int*)(accums + 256);        // N
  int*   idxx   = idxs + N;                    // N

  // output carving (floats, concatenated in tuple order)
  float* out   = (float*)d_out;
  float* emb_s = out;
  float* emb_x = emb_s + (size_t)N * DVQ;
  float* kidx  = emb_x + (size_t)N * DVQ;
  float* midx  = kidx + N;
  float* scal  = midx + N;

  k_prep<<<8, 256, 0, stream>>>(cb_s, cb_x, cns, cnx, cts, ctx, accums);

  k_spec_encode<<<N / 4, 256, 0, stream>>>(cubes, s_w1, s_b1, s_w2, s_b2,
                                           s_fc_w, s_fc_b, zes);
  k_spat_encode<<<N / 4, 256, 0, stream>>>(cubes, x_w1, x_b1, x_w2, x_b2,
                                           x_fc_w, x_fc_b, zex);

  k_vq_argmin<<<N / 64, 128, 0, stream>>>(zes, cb_s, cns, idxs);
  k_vq_argmin<<<N / 64, 128, 0, stream>>>(zex, cb_x, cnx, idxx);

  k_finish_vq<<<N / 4, 256, 0, stream>>>(zes, cb_s, idxs, emb_s, kidx, cts, accums + 0);
  k_finish_vq<<<N / 4, 256, 0, stream>>>(zex, cb_x, idxx, emb_x, midx, ctx, accums + 1);

  k_spec_decode<<<N / 4, 256, 0, stream>>>(emb_s, cubes, s_dec_fc_w, s_dec_fc_b,
                                           s_dec_w1, s_dec_b1, s_dec_w2, s_dec_b2,
                                           accums + 2);
  k_spat_decode<<<N / 4, 256, 0, stream>>>(emb_x, cubes, x_dec_fc_w, x_dec_fc_b,
                                           x_dec_w1, x_dec_b1, x_dec_w2, x_dec_b2,
                                           accums + 3);

  k_finalize<<<1, 256, 0, stream>>>(cts, ctx, accums, ret_recon, scal, N);
}